// GraphSAGE_26714696581620
// MI455X (gfx1250) — compile-verified
//
#include <hip/hip_runtime.h>
#include <hip/hip_bf16.h>

typedef float v2f __attribute__((ext_vector_type(2)));
typedef float v8f __attribute__((ext_vector_type(8)));

#define N_NODES 100000
#define N_EDGES 1200000

// ---------------------------------------------------------------------------
// Edge scatter: one wave32 per edge. Lanes cover 64 feature dims (2 each).
// Gather of feat[src*64 + lane] is a fully coalesced 256B read (L2-resident),
// accumulation uses no-return global_atomic_add_f32 into agg[dst].
// ---------------------------------------------------------------------------
__global__ __launch_bounds__(256) void sage_scatter(
    const float* __restrict__ feat, const long long* __restrict__ ei,
    const float* __restrict__ ew, float* __restrict__ agg,
    float* __restrict__ cnt, int n_edges, int do_cnt)
{
    int e = blockIdx.x * 8 + (threadIdx.x >> 5);   // 8 waves per block, 1 edge/wave
    if (e >= n_edges) return;
    int lane = threadIdx.x & 31;

    long long src = ei[e];
    long long dst = ei[n_edges + e];
    float w = ew[e];

    const float* xs = feat + (size_t)src * 64;
    float* ad       = agg  + (size_t)dst * 64;

    atomicAdd(&ad[lane],      xs[lane]      * w);
    atomicAdd(&ad[lane + 32], xs[lane + 32] * w);
    if (do_cnt && lane == 0) atomicAdd(&cnt[dst], 1.0f);
}

// ---------------------------------------------------------------------------
// Fused SAGE linear stage via V_WMMA_F32_16X16X4_F32:
//   out = [relu]( (agg / max(cnt,1)) @ Wl  +  xin @ Wr  + b )
// One wave per 16-node row tile; NT = number of 16-wide output tiles.
// A frag (16x4 f32, 2 VGPRs): lane m=l%16 holds rows, upper half-wave holds K+2.
// Weights are staged into LDS in *fragment order*: for each (kt, nt) k-step /
// n-tile, lane l's two B values (W[k0][col], W[k0+1][col]) are adjacent, so a
// B fragment is one aligned ds_load_b64 with no repacking movs and no bank
// conflicts (32 lanes x 2 dwords == 64 banks).
// C/D: v8f, VGPR r -> M = r + 8*(lane/16), N = lane%16.
// ---------------------------------------------------------------------------
template <int NT, bool RELU>
__global__ __launch_bounds__(256) void sage_gemm(
    const float* __restrict__ agg, const float* __restrict__ cnt,
    const float* __restrict__ xin, const float* __restrict__ Wl,
    const float* __restrict__ Wr, const float* __restrict__ bias,
    float* __restrict__ out, int n_nodes)
{
    constexpr int NOUT  = NT * 16;
    constexpr int NFRAG = 16 * NT;              // (kt, nt) fragment count
    __shared__ __align__(16) float sWl[NFRAG * 64];   // 64 floats per fragment
    __shared__ __align__(16) float sWr[NFRAG * 64];

    // Swizzle W[k][n] (row-major, K=64 x NOUT) into fragment order.
    for (int i = threadIdx.x; i < 64 * NOUT; i += 256) {
        int k = i / NOUT, n = i % NOUT;
        int kt = k >> 2, kr = k & 3;
        int kh = kr >> 1, ko = kr & 1;
        int nt = n >> 4,  m  = n & 15;
        int idx = ((kt * NT + nt) * 32 + kh * 16 + m) * 2 + ko;
        sWl[idx] = Wl[i];
        sWr[idx] = Wr[i];
    }
    __syncthreads();

    int wave = threadIdx.x >> 5;
    int lane = threadIdx.x & 31;
    int tile = blockIdx.x * 8 + wave;           // 16-node tile per wave
    int row0 = tile * 16;
    if (row0 >= n_nodes) return;

    int m  = lane & 15;                         // A row within tile
    int kh = lane >> 4;                         // half-wave K selector

    int node = row0 + m;
    float inv = 1.0f / fmaxf(cnt[node], 1.0f);

    const float* arow = agg + (size_t)node * 64;
    const float* xrow = xin + (size_t)node * 64;
    __builtin_prefetch(arow, 0, 0);
    __builtin_prefetch(xrow, 0, 0);

    v8f acc[NT];
    for (int t = 0; t < NT; ++t) acc[t] = (v8f)(0.0f);

#pragma unroll
    for (int kt = 0; kt < 16; ++kt) {           // K = 64 in steps of 4
        int k0 = kt * 4 + kh * 2;
        v2f aA, aX;
        aA.x = arow[k0] * inv;  aA.y = arow[k0 + 1] * inv;
        aX.x = xrow[k0];        aX.y = xrow[k0 + 1];
#pragma unroll
        for (int nt = 0; nt < NT; ++nt) {
            int fbase = ((kt * NT + nt) * 32 + lane) * 2;
            v2f bL = *(const v2f*)&sWl[fbase];  // single ds_load_b64
            v2f bR = *(const v2f*)&sWr[fbase];
            acc[nt] = __builtin_amdgcn_wmma_f32_16x16x4_f32(
                false, aA, false, bL, (short)0, acc[nt], false, false);
            acc[nt] = __builtin_amdgcn_wmma_f32_16x16x4_f32(
                false, aX, false, bR, (short)0, acc[nt], false, false);
        }
    }

#pragma unroll
    for (int nt = 0; nt < NT; ++nt) {
        int col = nt * 16 + m;
        float bv = bias[col];
#pragma unroll
        for (int r = 0; r < 8; ++r) {
            int mr = r + kh * 8;
            float v = acc[nt][r] + bv;
            if (RELU) v = fmaxf(v, 0.0f);
            out[(size_t)(row0 + mr) * NOUT + col] = v;
        }
    }
}

extern "C" void kernel_launch(void* const* d_in, const int* in_sizes, int n_in,
                              void* d_out, int out_size, void* d_ws, size_t ws_size,
                              hipStream_t stream) {
    const float*     x   = (const float*)d_in[0];
    const long long* ei  = (const long long*)d_in[1];
    const float*     ew  = (const float*)d_in[2];
    const float*     W1l = (const float*)d_in[3];
    const float*     b1  = (const float*)d_in[4];
    const float*     W1r = (const float*)d_in[5];
    const float*     W2l = (const float*)d_in[6];
    const float*     b2  = (const float*)d_in[7];
    const float*     W2r = (const float*)d_in[8];
    float* out = (float*)d_out;

    float* agg = (float*)d_ws;                       // N*64 f32
    float* cnt = agg + (size_t)N_NODES * 64;         // N    f32
    float* h   = cnt + N_NODES;                      // N*64 f32

    const int sblocks = (N_EDGES + 7) / 8;           // 1 edge per wave, 8 waves/block
    const int tiles   = N_NODES / 16;                // 6250 (exact)
    const int gblocks = (tiles + 7) / 8;             // 8 tiles per block

    // ---- Layer 1 ----
    hipMemsetAsync(agg, 0, sizeof(float) * ((size_t)N_NODES * 64 + N_NODES), stream);
    sage_scatter<<<sblocks, 256, 0, stream>>>(x, ei, ew, agg, cnt, N_EDGES, 1);
    sage_gemm<4, true><<<gblocks, 256, 0, stream>>>(agg, cnt, x, W1l, W1r, b1, h, N_NODES);

    // ---- Layer 2 (reuse agg buffer and degree counts) ----
    hipMemsetAsync(agg, 0, sizeof(float) * (size_t)N_NODES * 64, stream);
    sage_scatter<<<sblocks, 256, 0, stream>>>(h, ei, ew, agg, cnt, N_EDGES, 0);
    sage_gemm<1, false><<<gblocks, 256, 0, stream>>>(agg, cnt, h, W2l, W2r, b2, out, N_NODES);
}